// UnsupLabeledRisk_33698313404542
// MI455X (gfx1250) — compile-verified
//
#include <hip/hip_runtime.h>
#include <hip/hip_bf16.h>
#include <math.h>

// ---------------------------------------------------------------------------
// UnsupLabeledRisk: streaming reduction (min, max, sum, masked-sum, count)
// over 33.5M f32 + 33.5M i32 -> scalar. Memory-bound: 256 MB @ 23.3 TB/s
// ~ 11 us floor. Cross-lane sums use V_WMMA_F32_16X16X4_F32 (wave32).
// ---------------------------------------------------------------------------

typedef float v2f __attribute__((ext_vector_type(2)));
typedef float v8f __attribute__((ext_vector_type(8)));

#define NBLK 2048
#define NTHR 256
#define NWAVE (NTHR / 32)

// Full 32-lane sum via one WMMA: A[m][0]=p_m (lanes 0-15 vgpr0),
// A[m][2]=p_{m+16} (lanes 16-31 vgpr0), A[.][1]=A[.][3]=0, B=ones.
// => D[m][n] = p_m + p_{m+16} for every column n.
// Lane n<16 holds D[0..7][n] in its 8 acc VGPRs, lane n>=16 holds D[8..15][.].
// Sum of the 8 VGPRs + xor-shuffle by 16 => all lanes get the full sum.
__device__ __forceinline__ float wave_sum_f32(float v) {
    v2f a; a[0] = v;    a[1] = 0.0f;
    v2f b; b[0] = 1.0f; b[1] = 1.0f;
    v8f c = {};
    c = __builtin_amdgcn_wmma_f32_16x16x4_f32(false, a, false, b,
                                              (short)0, c, false, false);
    float t = ((c[0] + c[1]) + (c[2] + c[3])) + ((c[4] + c[5]) + (c[6] + c[7]));
    t += __shfl_xor(t, 16, 32);
    return t;
}

__device__ __forceinline__ int wave_sum_i32(int v) {
#pragma unroll
    for (int m = 16; m > 0; m >>= 1) v += __shfl_xor(v, m, 32);
    return v;
}

__device__ __forceinline__ float wave_min_f32(float v) {
#pragma unroll
    for (int m = 16; m > 0; m >>= 1) v = fminf(v, __shfl_xor(v, m, 32));
    return v;
}

__device__ __forceinline__ float wave_max_f32(float v) {
#pragma unroll
    for (int m = 16; m > 0; m >>= 1) v = fmaxf(v, __shfl_xor(v, m, 32));
    return v;
}

// Kernel 1: grid-stride streaming pass, one partial record per block.
// ws layout (floats): [0,NBLK) sum | [NBLK,2N) sum0 | [2N,3N) min | [3N,4N) max
// then NBLK ints for count0.
__global__ void __launch_bounds__(NTHR)
ulr_partial_kernel(const float* __restrict__ x, const int* __restrict__ y,
                   float* __restrict__ wsf, int* __restrict__ wsi, int n) {
    const int tid      = blockIdx.x * NTHR + threadIdx.x;
    const int nthreads = gridDim.x * NTHR;
    const int nv       = n >> 2;

    float s = 0.0f, s0 = 0.0f;
    int   c0 = 0;
    float mn = __builtin_inff(), mx = -__builtin_inff();

    const float4* __restrict__ x4 = (const float4*)x;
    const int4*   __restrict__ y4 = (const int4*)y;

    for (int i = tid; i < nv; i += nthreads) {
        float4 xv = x4[i];
        int4   yv = y4[i];
        s  += (xv.x + xv.y) + (xv.z + xv.w);
        s0 += (yv.x == 0 ? xv.x : 0.0f) + (yv.y == 0 ? xv.y : 0.0f)
            + (yv.z == 0 ? xv.z : 0.0f) + (yv.w == 0 ? xv.w : 0.0f);
        c0 += (yv.x == 0) + (yv.y == 0) + (yv.z == 0) + (yv.w == 0);
        mn  = fminf(mn, fminf(fminf(xv.x, xv.y), fminf(xv.z, xv.w)));
        mx  = fmaxf(mx, fmaxf(fmaxf(xv.x, xv.y), fmaxf(xv.z, xv.w)));
    }
    // scalar tail (n % 4 elements, handled by first few threads of the grid)
    const int base = nv << 2;
    if (tid < n - base) {
        float xv = x[base + tid];
        int   yv = y[base + tid];
        s += xv;
        if (yv == 0) { s0 += xv; ++c0; }
        mn = fminf(mn, xv);
        mx = fmaxf(mx, xv);
    }

    // wave-level reduction (EXEC all-ones here: full block, reconverged)
    s  = wave_sum_f32(s);
    s0 = wave_sum_f32(s0);
    c0 = wave_sum_i32(c0);
    mn = wave_min_f32(mn);
    mx = wave_max_f32(mx);

    __shared__ float sh_s[NWAVE], sh_s0[NWAVE], sh_mn[NWAVE], sh_mx[NWAVE];
    __shared__ int   sh_c[NWAVE];
    const int wave = threadIdx.x >> 5;
    const int lane = threadIdx.x & 31;
    if (lane == 0) {
        sh_s[wave] = s; sh_s0[wave] = s0; sh_c[wave] = c0;
        sh_mn[wave] = mn; sh_mx[wave] = mx;
    }
    __syncthreads();
    if (threadIdx.x == 0) {
        float bs = 0.0f, bs0 = 0.0f;
        float bmn = __builtin_inff(), bmx = -__builtin_inff();
        int bc = 0;
#pragma unroll
        for (int w = 0; w < NWAVE; ++w) {
            bs += sh_s[w]; bs0 += sh_s0[w]; bc += sh_c[w];
            bmn = fminf(bmn, sh_mn[w]); bmx = fmaxf(bmx, sh_mx[w]);
        }
        wsf[blockIdx.x]            = bs;
        wsf[NBLK + blockIdx.x]     = bs0;
        wsf[2 * NBLK + blockIdx.x] = bmn;
        wsf[3 * NBLK + blockIdx.x] = bmx;
        wsi[blockIdx.x]            = bc;
    }
}

// Kernel 2: one block reduces the NBLK partial records and emits the scalar.
__global__ void __launch_bounds__(NTHR)
ulr_final_kernel(const float* __restrict__ wsf, const int* __restrict__ wsi,
                 float* __restrict__ out, int n) {
    float s = 0.0f, s0 = 0.0f;
    float mn = __builtin_inff(), mx = -__builtin_inff();
    int c0 = 0;
#pragma unroll
    for (int i = threadIdx.x; i < NBLK; i += NTHR) {
        s  += wsf[i];
        s0 += wsf[NBLK + i];
        mn  = fminf(mn, wsf[2 * NBLK + i]);
        mx  = fmaxf(mx, wsf[3 * NBLK + i]);
        c0 += wsi[i];
    }

    s  = wave_sum_f32(s);
    s0 = wave_sum_f32(s0);
    c0 = wave_sum_i32(c0);
    mn = wave_min_f32(mn);
    mx = wave_max_f32(mx);

    __shared__ float sh_s[NWAVE], sh_s0[NWAVE], sh_mn[NWAVE], sh_mx[NWAVE];
    __shared__ int   sh_c[NWAVE];
    const int wave = threadIdx.x >> 5;
    const int lane = threadIdx.x & 31;
    if (lane == 0) {
        sh_s[wave] = s; sh_s0[wave] = s0; sh_c[wave] = c0;
        sh_mn[wave] = mn; sh_mx[wave] = mx;
    }
    __syncthreads();
    if (threadIdx.x == 0) {
        float bs = 0.0f, bs0 = 0.0f;
        float bmn = __builtin_inff(), bmx = -__builtin_inff();
        int bc = 0;
#pragma unroll
        for (int w = 0; w < NWAVE; ++w) {
            bs += sh_s[w]; bs0 += sh_s0[w]; bc += sh_c[w];
            bmn = fminf(bmn, sh_mn[w]); bmx = fmaxf(bmx, sh_mx[w]);
        }
        // Final algebra in f64: result is a difference of two nearly-equal
        // means built from ~1e8-scale intermediates (n0*xmin).
        const double range = (double)bmx - (double)bmn;
        const double n0    = (double)bc;
        const double n1    = (double)n - n0;
        const double S0n   = ((double)bs0 - n0 * (double)bmn) / range;
        const double S1n   = (((double)bs - (double)bs0) - n1 * (double)bmn) / range;
        const double m0    = S0n / n0;
        const double m1    = S1n / n1;
        out[0] = (float)(fmin(m0, m1) - fmax(m0, m1));
    }
}

extern "C" void kernel_launch(void* const* d_in, const int* in_sizes, int n_in,
                              void* d_out, int out_size, void* d_ws, size_t ws_size,
                              hipStream_t stream) {
    const float* x = (const float*)d_in[0];
    const int*   y = (const int*)d_in[1];
    float*       out = (float*)d_out;
    const int    n = in_sizes[0];

    float* wsf = (float*)d_ws;                 // 4*NBLK floats
    int*   wsi = (int*)(wsf + 4 * NBLK);       // NBLK ints  (total 40 KB)

    ulr_partial_kernel<<<NBLK, NTHR, 0, stream>>>(x, y, wsf, wsi, n);
    ulr_final_kernel<<<1, NTHR, 0, stream>>>(wsf, wsi, out, n);
}